// MaskedLinearMultivariate_37194416783526
// MI455X (gfx1250) — compile-verified
//
#include <hip/hip_runtime.h>

typedef __attribute__((ext_vector_type(2))) float v2f;
typedef __attribute__((ext_vector_type(8))) float v8f;

static constexpr int NT = 4096;   // N*T = 64*64
static constexpr int Dd = 32, Oo = 32, Gg = 8, Tt = 64;

// Grid: 512 blocks = (g*32+d)*2 + oh ; block = 256 threads (8 wave32).
// Each block computes xw-partial tile xw[g, 0:32, d, o0:o0+16] (no y yet):
//   part[blk][idx] = sum_k x[b,k] * weight[g,d,o,k] * mask[o, k%64]
// Waves split K = 4096 into 8 chunks of 512. Weight read exactly once chip-wide;
// x (512KB) is L2-resident, loaded directly in WMMA A-layout (no LDS staging).
__global__ __launch_bounds__(256) void mlm_main_kernel(
    const float* __restrict__ x,
    const float* __restrict__ w, const float* __restrict__ mask,
    float* __restrict__ part)
{
    __shared__ float pbuf[8 * 512];   // per-wave partial tiles, 16KB

    const int tid  = threadIdx.x;
    const int wave = tid >> 5;
    const int lane = tid & 31;
    const int blk  = blockIdx.x;
    const int oh   = blk & 1;
    const int d    = (blk >> 1) & 31;
    const int g    = blk >> 6;
    const int o0   = oh << 4;

    const int olane = lane & 15;          // B column (o) / A row (b) index
    const int khalf = (lane >> 4) << 1;   // 0 or 2: K sub-offset per WMMA layout

    // Mask pairs for this lane's o column, period 64 in K -> 16 v2f registers.
    v2f mreg[16];
    const float* mrow = mask + (o0 + olane) * Tt + khalf;
#pragma unroll
    for (int j = 0; j < 16; ++j)
        mreg[j] = *(const v2f*)(mrow + 4 * j);

    // B-matrix: masked weight row (contiguous in k). A-matrix: x rows directly.
    const float* wrow = w + (size_t)((g * Dd + d) * Oo + o0 + olane) * NT + khalf;
    const float* xa   = x + olane * NT + khalf;    // A tile0: b = 0..15
    const float* xb   = xa + 16 * NT;              // A tile1: b = 16..31

    v8f acc0 = {}, acc1 = {}, acc2 = {}, acc3 = {};
    const int kwave = wave << 9;                   // 8 waves x 512 K each

    for (int kc = 0; kc < 512; kc += 64) {
        const int k0 = kwave + kc;                 // multiple of 64 -> t phase 0
        const float* wp  = wrow + k0;
        const float* xap = xa + k0;
        const float* xbp = xb + k0;
#pragma unroll
        for (int j = 0; j < 16; j += 2) {
            const int kk = j << 2;
            v2f w0 = *(const v2f*)(wp + kk);
            v2f w1 = *(const v2f*)(wp + kk + 4);
            v2f b0 = w0 * mreg[j];
            v2f b1 = w1 * mreg[j + 1];
            v2f a00 = *(const v2f*)(xap + kk);
            v2f a10 = *(const v2f*)(xbp + kk);
            v2f a01 = *(const v2f*)(xap + kk + 4);
            v2f a11 = *(const v2f*)(xbp + kk + 4);
            acc0 = __builtin_amdgcn_wmma_f32_16x16x4_f32(false, a00, false, b0, (short)0, acc0, false, false);
            acc1 = __builtin_amdgcn_wmma_f32_16x16x4_f32(false, a10, false, b0, (short)0, acc1, false, false);
            acc2 = __builtin_amdgcn_wmma_f32_16x16x4_f32(false, a01, false, b1, (short)0, acc2, false, false);
            acc3 = __builtin_amdgcn_wmma_f32_16x16x4_f32(false, a11, false, b1, (short)0, acc3, false, false);
        }
        __builtin_prefetch(wp + 64, 0, 0);   // global_prefetch next weight chunk
    }
    acc0 += acc2;
    acc1 += acc3;

    // Per-wave partials -> LDS, cross-wave sum -> workspace (deterministic).
    float* pb = &pbuf[wave * 512];
#pragma unroll
    for (int r = 0; r < 8; ++r) {
        pb[r * 32 + lane]       = acc0[r];   // tile0: b = r + 8*(lane>=16)
        pb[256 + r * 32 + lane] = acc1[r];   // tile1: b = 16 + ...
    }
    __syncthreads();
    float* pblk = part + (size_t)blk * 512;
    for (int i = tid; i < 512; i += 256) {
        float s = 0.0f;
#pragma unroll
        for (int wv = 0; wv < 8; ++wv) s += pbuf[wv * 512 + i];
        pblk[i] = s;
    }
}

// out[b,d,o] = bias[d,o] + sum_g y[b,g] * xw[g,b,d,o]; fully overwrites d_out.
__global__ __launch_bounds__(256) void mlm_reduce_kernel(
    const float* __restrict__ part, const float* __restrict__ y,
    const float* __restrict__ bias, float* __restrict__ out)
{
    const int i = blockIdx.x * 256 + threadIdx.x;   // < 32768 = B*D*O
    const int o = i & 31;
    const int d = (i >> 5) & 31;
    const int b = i >> 10;
    const int oh = o >> 4, ol = o & 15;
    const int tile = b >> 4, rb = b & 15;
    const int lanei = ol + ((rb >> 3) << 4);        // invert D-matrix VGPR layout
    const int idx = tile * 256 + (rb & 7) * 32 + lanei;
    float s = bias[d * 32 + o];
#pragma unroll
    for (int g = 0; g < 8; ++g)
        s += y[b * 8 + g] * part[(size_t)((g * 32 + d) * 2 + oh) * 512 + idx];
    out[i] = s;
}

extern "C" void kernel_launch(void* const* d_in, const int* in_sizes, int n_in,
                              void* d_out, int out_size, void* d_ws, size_t ws_size,
                              hipStream_t stream) {
    const float* x    = (const float*)d_in[0];
    const float* y    = (const float*)d_in[1];
    const float* w    = (const float*)d_in[2];
    const float* bias = (const float*)d_in[3];
    const float* mask = (const float*)d_in[4];
    float* out  = (float*)d_out;
    float* part = (float*)d_ws;                    // 512 blocks * 512 f32 = 1 MB

    mlm_main_kernel<<<512, 256, 0, stream>>>(x, w, mask, part);
    mlm_reduce_kernel<<<128, 256, 0, stream>>>(part, y, bias, out);
}